// GATForSeqClsf_20134806684020
// MI455X (gfx1250) — compile-verified
//
#include <hip/hip_runtime.h>
#include <hip/hip_bf16.h>

typedef float v2f __attribute__((ext_vector_type(2)));
typedef float v8f __attribute__((ext_vector_type(8)));

#define NN 8192
#define KW 8   // waves per block in the fused GAT kernel (K-split of src dim)

// ---------------------------------------------------------------- embedding
__global__ void __launch_bounds__(256) k_embed(const int* __restrict__ wid,
                                               const float* __restrict__ emb,
                                               float* __restrict__ h) {
  int t = blockIdx.x * 256 + threadIdx.x;       // N*64 threads
  int n = t >> 6, d = t & 63;
  h[t] = emb[(size_t)wid[n] * 64 + d];
}

// ------------------------------------------------- GEMM: C[8192,Nc] = A[8192,64] @ W[64,Nc]
// one wave per 16x16 C tile; compile-time Nc -> all strides become immediates.
template <int Nc>
__global__ void __launch_bounds__(32) k_gemm(const float* __restrict__ A,
                                             const float* __restrict__ W,
                                             float* __restrict__ C) {
  const int lane = threadIdx.x & 31;
  const int m0 = blockIdx.x << 4;
  const int n0 = blockIdx.y << 4;
  const int ml = lane & 15;
  const int kb = (lane >> 4) << 1;              // K pair base: 0 or 2 per half-wave
  const int rhi = (lane >> 4) << 3;             // C row offset 0 / 8 per half-wave

  const float* __restrict__ Ap = A + (size_t)(m0 + ml) * 64 + kb; // 8B aligned
  const float* __restrict__ Wp = W + (size_t)kb * Nc + n0 + ml;
  float* __restrict__ Cp = C + (size_t)(m0 + rhi) * Nc + n0 + ml;

  v8f c = {0.f,0.f,0.f,0.f,0.f,0.f,0.f,0.f};
#pragma unroll
  for (int k = 0; k < 64; k += 4) {
    v2f a = *(const v2f*)(Ap + k);              // A[m, k+kb], A[m, k+kb+1]
    v2f b;
    b.x = Wp[(size_t)k * Nc];                   // W[k+kb,   n]
    b.y = Wp[(size_t)(k + 1) * Nc];             // W[k+kb+1, n]
    c = __builtin_amdgcn_wmma_f32_16x16x4_f32(false, a, false, b, (short)0, c,
                                              false, false);
  }
#pragma unroll
  for (int r = 0; r < 8; ++r)
    Cp[(size_t)r * Nc] = c[r];
}

// ---------------------------------------------- attention scores ss/sd [4, N]
template <int HD>
__global__ void __launch_bounds__(256) k_scores(const float* __restrict__ WH,
                                                const float* __restrict__ asrc,
                                                const float* __restrict__ adst,
                                                float* __restrict__ ss,
                                                float* __restrict__ sd) {
  int t = blockIdx.x * 256 + threadIdx.x;       // N*4 threads
  int n = t >> 2, h = t & 3;
  const float* w = WH + (size_t)n * (4 * HD) + h * HD;
  float s0 = 0.f, s1 = 0.f;
#pragma unroll
  for (int d = 0; d < HD; ++d) {
    float v = w[d];
    s0 += v * asrc[h * HD + d];
    s1 += v * adst[h * HD + d];
  }
  ss[h * NN + n] = s0;
  sd[h * NN + n] = s1;
}

// ------------------------------------- fused masked-softmax GAT layer (hd=16)
// Block = KW waves. Each wave accumulates a disjoint NN/KW-src range of the
// softmax numerator (WMMA f32 16x16x4) + denominator for all 4 heads over a
// 16-dst tile; LDS combine, then wave w<4 finalizes head w (norm/elu/resid).
__global__ void __launch_bounds__(32 * KW) k_gat_mid(
    const float* __restrict__ adj, const float* __restrict__ WH,
    const float* __restrict__ ss, const float* __restrict__ sd,
    const float* __restrict__ hin, float* __restrict__ hout) {
  const int lane = threadIdx.x & 31;
  const int wv = threadIdx.x >> 5;              // 0..KW-1
  const int j0 = blockIdx.x << 4;
  const int jl = lane & 15;
  const int kb = (lane >> 4) << 1;
  const int rhi = (lane >> 4) << 3;

  __shared__ float s_acc[KW][4][8][32];
  __shared__ float s_den[KW][4][32];

  v8f acc[4];
  float den[4];
  float sdv[4];
#pragma unroll
  for (int h = 0; h < 4; ++h) {
    acc[h] = (v8f){0.f,0.f,0.f,0.f,0.f,0.f,0.f,0.f};
    den[h] = 0.f;
    sdv[h] = sd[h * NN + j0 + jl];
  }

  const int ibeg = wv * (NN / KW);
  const int iend = ibeg + (NN / KW);

  // loop-carried base pointers; all per-tile accesses use constant offsets
  const float* __restrict__ ap = adj + (size_t)(ibeg + kb) * NN + j0 + jl;
  const float* __restrict__ wp = WH + (size_t)(ibeg + kb) * 64 + jl;
  const float* __restrict__ sp = ss + ibeg + kb;

  for (int i0 = ibeg; i0 < iend; i0 += 16) {
    // speculative prefetch of next adjacency tile (OOB at range end is
    // silently dropped by hardware for speculative prefetches)
    __builtin_prefetch(ap + (size_t)16 * NN, 0, 1);
    float am[8];
    bool anyv = false;
#pragma unroll
    for (int c = 0; c < 4; ++c) {
      am[2 * c]     = ap[(size_t)(4 * c) * NN];
      am[2 * c + 1] = ap[(size_t)(4 * c + 1) * NN];
      anyv = anyv || (am[2 * c] > 0.f) || (am[2 * c + 1] > 0.f);
    }
    if (__ballot(anyv) != 0ull) {               // skip empty 16x16 adj tiles
#pragma unroll
      for (int h = 0; h < 4; ++h) {
#pragma unroll
        for (int c = 0; c < 4; ++c) {
          float e0 = sp[h * NN + 4 * c]     + sdv[h];
          float e1 = sp[h * NN + 4 * c + 1] + sdv[h];
          e0 = (e0 > 0.f) ? e0 : 0.2f * e0;     // leaky_relu(0.2)
          e1 = (e1 > 0.f) ? e1 : 0.2f * e1;
          v2f a, b;
          a.x = (am[2 * c]     > 0.f) ? __expf(e0) : 0.f;
          a.y = (am[2 * c + 1] > 0.f) ? __expf(e1) : 0.f;
          den[h] += a.x + a.y;
          b.x = wp[(4 * c) * 64     + h * 16];
          b.y = wp[(4 * c + 1) * 64 + h * 16];
          acc[h] = __builtin_amdgcn_wmma_f32_16x16x4_f32(
              false, a, false, b, (short)0, acc[h], false, false);
        }
      }
    }
    ap += (size_t)16 * NN;
    wp += 16 * 64;
    sp += 16;
  }

  // publish partials
#pragma unroll
  for (int h = 0; h < 4; ++h) {
#pragma unroll
    for (int r = 0; r < 8; ++r) s_acc[wv][h][r][lane] = acc[h][r];
    s_den[wv][h][lane] = den[h];
  }
  __syncthreads();

  // waves 0..3 finalize heads 0..3
  if (wv < 4) {
    const int h = wv;
    float dtot = 0.f;
#pragma unroll
    for (int p = 0; p < KW; ++p) dtot += s_den[p][h][lane];
    dtot += __shfl_xor(dtot, 16, 32);           // lane l: den for dst j0+(l&15)
#pragma unroll
    for (int r = 0; r < 8; ++r) {
      float v = 0.f;
#pragma unroll
      for (int p = 0; p < KW; ++p) v += s_acc[p][h][r][lane];
      const float dn = __shfl(dtot, r + rhi, 32);
      v = v / dn;
      v = (v > 0.f) ? v : (__expf(v) - 1.f);    // elu
      const size_t idx = (size_t)(j0 + r + rhi) * 64 + h * 16 + jl;
      hout[idx] = v + hin[idx];                 // residual
    }
  }
}

// ------------------- last GAT layer evaluated only at the 32 cls dst nodes
__global__ void __launch_bounds__(256) k_gat_cls(const float* __restrict__ adj,
                                                 const float* __restrict__ WH3,
                                                 const float* __restrict__ ss,
                                                 const float* __restrict__ sd,
                                                 const int* __restrict__ cls,
                                                 float* __restrict__ hcls) {
  __shared__ float red[256];
  __shared__ float pbuf[4][256];
  __shared__ float outs[256];
  __shared__ float densh[4];
  const int t = threadIdx.x;
  const int dst = cls[blockIdx.x];
  float sdv[4];
#pragma unroll
  for (int h = 0; h < 4; ++h) sdv[h] = sd[h * NN + dst];

  // pass 1: softmax denominators per head
  float dacc[4] = {0.f, 0.f, 0.f, 0.f};
  for (int i = t; i < NN; i += 256) {
    const float m = adj[(size_t)i * NN + dst];
    if (m > 0.f) {
#pragma unroll
      for (int h = 0; h < 4; ++h) {
        float e = ss[h * NN + i] + sdv[h];
        e = (e > 0.f) ? e : 0.2f * e;
        dacc[h] += __expf(e);
      }
    }
  }
  for (int h = 0; h < 4; ++h) {
    red[t] = dacc[h];
    __syncthreads();
    for (int s = 128; s > 0; s >>= 1) {
      if (t < s) red[t] += red[t + s];
      __syncthreads();
    }
    if (t == 0) densh[h] = red[0];
    __syncthreads();
  }

  // pass 2: thread t owns output dim t (head t/64, dim t%64)
  const int hh = t >> 6;
  float accv = 0.f;
  for (int i0 = 0; i0 < NN; i0 += 256) {
    const int i = i0 + t;
    const float m = adj[(size_t)i * NN + dst];
#pragma unroll
    for (int h = 0; h < 4; ++h) {
      float p = 0.f;
      if (m > 0.f) {
        float e = ss[h * NN + i] + sdv[h];
        e = (e > 0.f) ? e : 0.2f * e;
        p = __expf(e) / densh[h];
      }
      pbuf[h][t] = p;
    }
    __syncthreads();
    for (int ii = 0; ii < 256; ++ii)
      accv += pbuf[hh][ii] * WH3[(size_t)(i0 + ii) * 256 + t];
    __syncthreads();
  }
  outs[t] = accv;
  __syncthreads();
  if (t < 64)                                   // mean over heads
    hcls[blockIdx.x * 64 + t] =
        0.25f * (outs[t] + outs[64 + t] + outs[128 + t] + outs[192 + t]);
}

// ----------------------------- logits + NLL loss; d_out = [logits(64), loss]
__global__ void __launch_bounds__(32) k_final(const float* __restrict__ hcls,
                                              const float* __restrict__ lw,
                                              const float* __restrict__ lb,
                                              const int* __restrict__ y,
                                              float* __restrict__ out) {
  __shared__ float ls[32];
  const int b = threadIdx.x;                    // 32 graphs
  float l0 = lb[0], l1 = lb[1];
  for (int d = 0; d < 64; ++d) {
    float v = hcls[b * 64 + d];
    l0 += v * lw[d * 2 + 0];
    l1 += v * lw[d * 2 + 1];
  }
  out[2 * b + 0] = l0;
  out[2 * b + 1] = l1;
  float m = fmaxf(l0, l1);
  float lse = m + __logf(__expf(l0 - m) + __expf(l1 - m));
  float ly = ((y[b] == 0) ? l0 : l1) - lse;
  ls[b] = -ly;
  __syncthreads();
  if (b == 0) {
    float s = 0.f;
    for (int k = 0; k < 32; ++k) s += ls[k];
    out[64] = s / 32.f;
  }
}

extern "C" void kernel_launch(void* const* d_in, const int* in_sizes, int n_in,
                              void* d_out, int out_size, void* d_ws,
                              size_t ws_size, hipStream_t stream) {
  const int*   word_ids = (const int*)d_in[0];
  const float* adj      = (const float*)d_in[1];
  // d_in[2] = edge_type (unused by the reference forward pass)
  const int*   cls_node = (const int*)d_in[3];
  const int*   y        = (const int*)d_in[4];
  const float* emb      = (const float*)d_in[5];
  const float* Ws       = (const float*)d_in[6];   // [2,64,64]
  const float* a_src    = (const float*)d_in[7];   // [2,4,16]
  const float* a_dst    = (const float*)d_in[8];   // [2,4,16]
  const float* W_last   = (const float*)d_in[9];   // [64,256]
  const float* a_src_l  = (const float*)d_in[10];  // [4,64]
  const float* a_dst_l  = (const float*)d_in[11];  // [4,64]
  const float* lin_w    = (const float*)d_in[12];  // [64,2]
  const float* lin_b    = (const float*)d_in[13];  // [2]

  float* wsf  = (float*)d_ws;
  float* H0   = wsf;                    // 8192*64
  float* H1   = H0 + (size_t)NN * 64;
  float* WH   = H1 + (size_t)NN * 64;   // 8192*256 (covers both mid WH and WH3)
  float* SS   = WH + (size_t)NN * 256;  // 4*8192
  float* SD   = SS + 4 * NN;            // 4*8192
  float* HCLS = SD + 4 * NN;            // 32*64

  k_embed<<<dim3(NN * 64 / 256), dim3(256), 0, stream>>>(word_ids, emb, H0);

  const float* hc = H0;
  float* hn = H1;
  for (int l = 0; l < 2; ++l) {
    k_gemm<64><<<dim3(NN / 16, 4), dim3(32), 0, stream>>>(hc, Ws + l * 64 * 64,
                                                          WH);
    k_scores<16><<<dim3(NN * 4 / 256), dim3(256), 0, stream>>>(
        WH, a_src + l * 64, a_dst + l * 64, SS, SD);
    k_gat_mid<<<dim3(NN / 16), dim3(32 * KW), 0, stream>>>(adj, WH, SS, SD, hc,
                                                           hn);
    const float* tmp = hc; hc = hn; hn = (float*)tmp;
  }

  k_gemm<256><<<dim3(NN / 16, 16), dim3(32), 0, stream>>>(hc, W_last, WH);
  k_scores<64><<<dim3(NN * 4 / 256), dim3(256), 0, stream>>>(WH, a_src_l,
                                                             a_dst_l, SS, SD);
  k_gat_cls<<<dim3(32), dim3(256), 0, stream>>>(adj, WH, SS, SD, cls_node, HCLS);
  k_final<<<dim3(1), dim3(32), 0, stream>>>(HCLS, lin_w, lin_b, y,
                                            (float*)d_out);
}